// GraphSAGE_35510789603342
// MI455X (gfx1250) — compile-verified
//
#include <hip/hip_runtime.h>

// GraphSAGE (3-layer, mean aggregation, residual) for MI455X / gfx1250.
// - Aggregation: L2-resident f32 atomics (x/agg both << 192MB L2).
// - GEMMs: V_WMMA_F32_16X16X4_F32, weights staged in LDS (padded stride 132
//   floats -> conflict-free b64 fragment reads). Two-pass K reduction with
//   double-buffered fragments so each k-step is: 8 ds loads in flight while
//   8 WMMAs retire (single dscnt wait per step, no per-WMMA stalls).
// - ReLU fused into the GEMM epilogue.

#define NN 50000
#define NE 800000
#define DD 128
#define NT (NN / 16)            // 3125 full 16-node tiles (50000 % 16 == 0)
#define FEAT (NN * DD)
#define WSTRIDE 132             // padded LDS row stride (floats): bank step 4
#define WOFF (DD * WSTRIDE)     // offset of Wr inside LDS (floats)

typedef float v2f __attribute__((ext_vector_type(2)));
typedef float v8f __attribute__((ext_vector_type(8)));

__device__ __forceinline__ v8f wmma4(v2f a, v2f b, v8f c) {
  // D = A(16x4 f32) * B(4x16 f32) + C(16x16 f32)
  return __builtin_amdgcn_wmma_f32_16x16x4_f32(false, a, false, b, (short)0, c,
                                               false, false);
}

__global__ void k_zero(float4* __restrict__ p, int n4) {
  int i = blockIdx.x * blockDim.x + threadIdx.x;
  int stride = gridDim.x * blockDim.x;
  float4 z = {0.f, 0.f, 0.f, 0.f};
  for (; i < n4; i += stride) p[i] = z;
}

__global__ void k_deg(const int* __restrict__ row, float* __restrict__ deg) {
  int e = blockIdx.x * blockDim.x + threadIdx.x;
  if (e < NE) atomicAdd(&deg[row[e]], 1.0f);
}

// One thread per (edge, 4-float chunk): float4 gather + 4 f32 atomic adds.
__global__ void k_scatter(const float* __restrict__ xin,
                          const int* __restrict__ row,
                          const int* __restrict__ col,
                          float* __restrict__ agg) {
  long tid = (long)blockIdx.x * blockDim.x + threadIdx.x;
  int e = (int)(tid >> 5);
  if (e >= NE) return;
  int c = ((int)tid & 31) << 2;
  int s = col[e];
  int d = row[e];
  const float4 v = *(const float4*)(xin + (size_t)s * DD + c);
  float* dst = agg + (size_t)d * DD + c;
  atomicAdd(dst + 0, v.x);
  atomicAdd(dst + 1, v.y);
  atomicAdd(dst + 2, v.z);
  atomicAdd(dst + 3, v.w);
}

// Fused SAGE layer:
//   y[i,:]    = (agg[i,:]*invdeg[i]) @ Wl^T + h[i,:] @ Wr^T + bias (+ resid)
//   hout[i,:] = relu(y[i,:])   (optional, for the next layer)
// One wave per 16-node tile; 8 waves (tiles) per 256-thread block; weights
// live in LDS (staged once per block). A-frag (ISA 7.12.2, 32-bit A 16x4):
// lanes 0-15 rows m0+L with K,K+1; lanes 16-31 same rows with K+2,K+3.
// B mirrors over N: lane holds W[(n0+lm)][K + 2*half + {0,1}] (B = W^T).
__global__ void k_sage_gemm(const float* __restrict__ agg,
                            const float* __restrict__ deg,
                            const float* __restrict__ h,
                            const float* __restrict__ Wl,
                            const float* __restrict__ Wr,
                            const float* __restrict__ bias,
                            const float* __restrict__ resid,  // may be null
                            float* __restrict__ out,
                            float* __restrict__ hout) {       // may be null
  __shared__ float wlds[2 * DD * WSTRIDE];  // 132 KB: Wl then Wr, stride 132

  // Stage both weight matrices into LDS (b128 global loads -> b128 ds stores).
  for (int i = threadIdx.x; i < (DD * DD) / 4; i += 256) {
    int o = i >> 5;             // output row 0..127
    int kc = (i & 31) << 2;     // k chunk 0,4,...,124
    float4 a = *(const float4*)(Wl + (size_t)o * DD + kc);
    float4 b = *(const float4*)(Wr + (size_t)o * DD + kc);
    *(float4*)(&wlds[o * WSTRIDE + kc]) = a;
    *(float4*)(&wlds[WOFF + o * WSTRIDE + kc]) = b;
  }
  __syncthreads();

  const int nw = blockDim.x >> 5;                 // waves per block (8)
  const int wv = threadIdx.x >> 5;
  const int lane = threadIdx.x & 31;
  const int half = lane >> 4;                     // 0: K,K+1  1: K+2,K+3
  const int lm = lane & 15;
  const int tstride = gridDim.x * nw;

  for (int tile = blockIdx.x * nw + wv; tile < NT; tile += tstride) {
    const int m0 = tile << 4;
    // Prefetch next tile's A rows into L2 (global_prefetch_b8).
    if (tile + tstride < NT) {
      const size_t nb = (size_t)((tile + tstride) * 16 + lm) * DD;
      __builtin_prefetch(agg + nb, 0, 1);
      __builtin_prefetch(h + nb, 0, 1);
    }

    const float sc = 1.0f / fmaxf(deg[m0 + lm], 1.0f);
    const float* arow = agg + (size_t)(m0 + lm) * DD + 2 * half;
    const float* hrow = h + (size_t)(m0 + lm) * DD + 2 * half;
    const float* blbase = &wlds[(size_t)lm * WSTRIDE + 2 * half];
    const float* brbase = &wlds[WOFF + (size_t)lm * WSTRIDE + 2 * half];

    v8f acc[8] = {};  // 8 n-tiles x 16x16 f32

    // Two passes over K: pass 0 = neighbor-mean @ Wl^T (scaled by invdeg),
    // pass 1 = self @ Wr^T. Only 8 live B-fragments per pass -> the
    // double-buffer below fits in registers and pipelines cleanly.
#pragma unroll 1
    for (int pass = 0; pass < 2; ++pass) {
      const float* aptr = pass ? hrow : arow;
      const float* bb = pass ? brbase : blbase;
      const float scale = pass ? 1.0f : sc;

      v2f cur[8], nxt[8];
#pragma unroll
      for (int t = 0; t < 8; ++t)
        cur[t] = *(const v2f*)(bb + (size_t)(t << 4) * WSTRIDE);
      float2 a2 = *(const float2*)(aptr);

#pragma unroll 2
      for (int k = 0; k < 32; ++k) {
        const int K1 = ((k + 1) & 31) << 2;  // wrap: keep loads unconditional
        // Issue next k-step's fragment loads (8x ds_load_b64 + 1 global b64)
        // before consuming the current ones.
#pragma unroll
        for (int t = 0; t < 8; ++t)
          nxt[t] = *(const v2f*)(bb + (size_t)(t << 4) * WSTRIDE + K1);
        float2 a2n = *(const float2*)(aptr + K1);

        v2f va;
        va.x = a2.x * scale;
        va.y = a2.y * scale;
#pragma unroll
        for (int t = 0; t < 8; ++t) acc[t] = wmma4(va, cur[t], acc[t]);

#pragma unroll
        for (int t = 0; t < 8; ++t) cur[t] = nxt[t];
        a2 = a2n;
      }
    }

    // D layout (ISA 7.12.2): lane<16 -> N=lane, VGPR r -> M=r;
    //                        lane>=16 -> N=lane-16, VGPR r -> M=8+r.
#pragma unroll
    for (int t = 0; t < 8; ++t) {
      const int n0 = t << 4;
      const float bb = bias[n0 + lm];
#pragma unroll
      for (int r = 0; r < 8; ++r) {
        const size_t idx = (size_t)(m0 + r + (half << 3)) * DD + n0 + lm;
        float v = acc[t][r] + bb;
        if (resid) v += resid[idx];
        out[idx] = v;
        if (hout) hout[idx] = fmaxf(v, 0.0f);
      }
    }
  }
}

extern "C" void kernel_launch(void* const* d_in, const int* in_sizes, int n_in,
                              void* d_out, int out_size, void* d_ws,
                              size_t ws_size, hipStream_t stream) {
  const float* x = (const float*)d_in[0];
  const int* erow = (const int*)d_in[1];
  const int* ecol = (const int*)d_in[2];
  const float* Wl0 = (const float*)d_in[3];
  const float* Wr0 = (const float*)d_in[4];
  const float* b0 = (const float*)d_in[5];
  const float* Wl1 = (const float*)d_in[6];
  const float* Wr1 = (const float*)d_in[7];
  const float* b1 = (const float*)d_in[8];
  const float* Wl2 = (const float*)d_in[9];
  const float* Wr2 = (const float*)d_in[10];
  const float* b2 = (const float*)d_in[11];
  float* out = (float*)d_out;

  float* ws = (float*)d_ws;
  float* deg = ws;                       // NN floats (NN % 4 == 0)
  float* agg = ws + NN;                  // NN*DD
  float* h = agg + (size_t)FEAT;         // NN*DD
  float* ycur = h + (size_t)FEAT;        // NN*DD

  dim3 b256(256);
  const int degBlocks = (NE + 255) / 256;
  const int scatterBlocks = (int)(((long)NE * 32 + 255) / 256);
  const int gemmBlocks = (NT + 7) / 8;   // 8 waves (tiles) per 256-thr block

  // Degrees (layer-invariant).
  k_zero<<<dim3(64), b256, 0, stream>>>((float4*)deg, NN / 4);
  k_deg<<<dim3(degBlocks), b256, 0, stream>>>(erow, deg);

  // Layer 0: agg = scatter(x); y1 = sage(agg, x); h = relu(y1)
  k_zero<<<dim3(2048), b256, 0, stream>>>((float4*)agg, FEAT / 4);
  k_scatter<<<dim3(scatterBlocks), b256, 0, stream>>>(x, erow, ecol, agg);
  k_sage_gemm<<<dim3(gemmBlocks), b256, 0, stream>>>(agg, deg, x, Wl0, Wr0, b0,
                                                     nullptr, ycur, h);

  // Layer 1: y2 = sage(scatter(h), h) + y1; h = relu(y2)
  k_zero<<<dim3(2048), b256, 0, stream>>>((float4*)agg, FEAT / 4);
  k_scatter<<<dim3(scatterBlocks), b256, 0, stream>>>(h, erow, ecol, agg);
  k_sage_gemm<<<dim3(gemmBlocks), b256, 0, stream>>>(agg, deg, h, Wl1, Wr1, b1,
                                                     ycur, ycur, h);

  // Layer 2: out = sage(scatter(h), h) + y2
  k_zero<<<dim3(2048), b256, 0, stream>>>((float4*)agg, FEAT / 4);
  k_scatter<<<dim3(scatterBlocks), b256, 0, stream>>>(h, erow, ecol, agg);
  k_sage_gemm<<<dim3(gemmBlocks), b256, 0, stream>>>(agg, deg, h, Wl2, Wr2, b2,
                                                     ycur, out, nullptr);
}